// SetAbstractionMSG_48326972014678
// MI455X (gfx1250) — compile-verified
//
#include <hip/hip_runtime.h>
#include <hip/hip_bf16.h>
#include <stdint.h>

typedef float v2f __attribute__((ext_vector_type(2)));
typedef float v8f __attribute__((ext_vector_type(8)));

static __device__ __forceinline__ v8f wmma4(v2f a, v2f b, v8f c) {
  // D(16x16,f32) = A(16x4,f32) x B(4x16,f32) + C
  return __builtin_amdgcn_wmma_f32_16x16x4_f32(false, a, false, b, (short)0, c, false, false);
}

#define NPTS 4096
#define NOUT 1024
#define OUTC 323

// ---------------------------------------------------------------------------
// Kernel 1: Farthest-point sampling. One block per batch, serial 1024 steps.
// ---------------------------------------------------------------------------
#define FPS_T 256
#define FPS_PER (NPTS / FPS_T)

__global__ __launch_bounds__(FPS_T) void fps_kernel(
    const float* __restrict__ x, const int* __restrict__ start_idx,
    float* __restrict__ centroids, float* __restrict__ out)
{
  __shared__ float sx[NPTS * 3];
  __shared__ float rv[FPS_T / 32];
  __shared__ int   ri[FPS_T / 32];
  __shared__ int   s_last;
  const int b   = blockIdx.x;
  const int tid = threadIdx.x;
  const int lane = tid & 31;
  const int wid  = tid >> 5;
  const float* xb = x + (size_t)b * NPTS * 3;
  for (int i = tid; i < NPTS * 3; i += FPS_T) sx[i] = xb[i];
  __syncthreads();

  float md[FPS_PER];
#pragma unroll
  for (int j = 0; j < FPS_PER; ++j) md[j] = __builtin_inff();

  int last = start_idx[b];
  for (int k = 0; k < NOUT; ++k) {
    float px = sx[last * 3 + 0], py = sx[last * 3 + 1], pz = sx[last * 3 + 2];
    if (tid == 0) {
      size_t cb = (size_t)b * NOUT + k;
      centroids[cb * 3 + 0] = px; centroids[cb * 3 + 1] = py; centroids[cb * 3 + 2] = pz;
      float* op = out + cb * OUTC;
      op[0] = px; op[1] = py; op[2] = pz;
    }
    float best = -__builtin_inff(); int bidx = 0;
#pragma unroll
    for (int j = 0; j < FPS_PER; ++j) {
      int i = tid + j * FPS_T;   // ascending index per thread
      float dx = __fsub_rn(sx[i * 3 + 0], px);
      float dy = __fsub_rn(sx[i * 3 + 1], py);
      float dz = __fsub_rn(sx[i * 3 + 2], pz);
      float d  = __fadd_rn(__fadd_rn(__fmul_rn(dx, dx), __fmul_rn(dy, dy)), __fmul_rn(dz, dz));
      float m  = fminf(md[j], d);
      md[j] = m;
      if (m > best) { best = m; bidx = i; }   // strict > keeps smallest index
    }
#pragma unroll
    for (int off = 16; off >= 1; off >>= 1) {
      float ov = __shfl_xor(best, off, 32);
      int   oi = __shfl_xor(bidx, off, 32);
      if (ov > best || (ov == best && oi < bidx)) { best = ov; bidx = oi; }
    }
    if (lane == 0) { rv[wid] = best; ri[wid] = bidx; }
    __syncthreads();
    if (tid == 0) {
      float bb = rv[0]; int bi = ri[0];
      for (int w = 1; w < FPS_T / 32; ++w)
        if (rv[w] > bb || (rv[w] == bb && ri[w] < bi)) { bb = rv[w]; bi = ri[w]; }
      s_last = bi;
    }
    __syncthreads();
    last = s_last;
  }
}

// ---------------------------------------------------------------------------
// Kernel 2: Ball query + grouping. One wave per centroid; ordered compaction
// via ballot keeps the first K in-radius indices (ascending); ref-style
// padding with the first hit (or point 0 when no hit).
// ---------------------------------------------------------------------------
__global__ __launch_bounds__(256) void ballq_kernel(
    const float* __restrict__ x, const float* __restrict__ centroids,
    float* __restrict__ g, int K, float r2)
{
  const int lane = threadIdx.x & 31;
  const int w = blockIdx.x * 8 + (threadIdx.x >> 5);   // 0..8191
  const int b = w >> 10;
  const float cx = centroids[w * 3 + 0];
  const float cy = centroids[w * 3 + 1];
  const float cz = centroids[w * 3 + 2];
  const float* xb = x + (size_t)b * NPTS * 3;
  float* gw = g + (size_t)w * K * 3;
  int cnt = 0;
  bool have = false;
  float fx = 0.f, fy = 0.f, fz = 0.f;
  for (int base = 0; base < NPTS; base += 32) {
    int i = base + lane;
    float px = xb[i * 3 + 0], py = xb[i * 3 + 1], pz = xb[i * 3 + 2];
    float dx = __fsub_rn(cx, px), dy = __fsub_rn(cy, py), dz = __fsub_rn(cz, pz);
    float d2 = __fadd_rn(__fadd_rn(__fmul_rn(dx, dx), __fmul_rn(dy, dy)), __fmul_rn(dz, dz));
    bool pred = (d2 <= r2);
    unsigned mask = __builtin_amdgcn_ballot_w32(pred);
    float rx = __fsub_rn(px, cx), ry = __fsub_rn(py, cy), rz = __fsub_rn(pz, cz);
    int rank = __builtin_popcount(mask & ((1u << lane) - 1u));
    int slot = cnt + rank;
    if (pred && slot < K) {
      gw[slot * 3 + 0] = rx; gw[slot * 3 + 1] = ry; gw[slot * 3 + 2] = rz;
    }
    if (!have && mask) {
      int sl = __ffs(mask) - 1;
      fx = __shfl(rx, sl, 32); fy = __shfl(ry, sl, 32); fz = __shfl(rz, sl, 32);
      have = true;
    }
    cnt += __builtin_popcount(mask);
    if (cnt >= K) break;          // uniform across the wave
  }
  if (!have) {                    // no point in radius -> index 0
    fx = __fsub_rn(xb[0], cx); fy = __fsub_rn(xb[1], cy); fz = __fsub_rn(xb[2], cz);
  }
  for (int s = cnt + lane; s < K; s += 32) {
    gw[s * 3 + 0] = fx; gw[s * 3 + 1] = fy; gw[s * 3 + 2] = fz;
  }
}

// ---------------------------------------------------------------------------
// Kernel 3: 3-layer pointwise MLP + maxpool over K, WMMA f32 16x16x4.
// Wave per centroid, 8 waves/block. Weights staged TRANSPOSED in LDS
// (WT[n][k], stride C+4 -> one ds_load_b64 per B fragment, bank-conflict
// free). Per-wave 16x96 staging tile (stride 100) re-layouts D-tiles into
// A-operands; same-wave LDS ordering (DScnt, in-order) makes block barriers
// unnecessary inside the row-tile loop.
// ---------------------------------------------------------------------------
template <int K, int C1, int C2, int C3>
__global__ __launch_bounds__(256) void mlp_kernel(
    const float* __restrict__ g,
    const float* __restrict__ W1, const float* __restrict__ B1,
    const float* __restrict__ W2, const float* __restrict__ B2,
    const float* __restrict__ W3, const float* __restrict__ B3,
    float* __restrict__ out, int chanoff)
{
  extern __shared__ float smem[];
  constexpr int ST1 = 6;        // WT1: [C1][4 pad 6]
  constexpr int ST2 = C1 + 4;   // WT2: [C2][C1 pad]
  constexpr int ST3 = C2 + 4;   // WT3: [C3][C2 pad]
  constexpr int oB1 = C1 * ST1;
  constexpr int oW2 = oB1 + C1;
  constexpr int oB2 = oW2 + C2 * ST2;
  constexpr int oW3 = oB2 + C2;
  constexpr int oB3 = oW3 + C3 * ST3;
  constexpr int oST = oB3 + C3;
  constexpr int STR = 100;      // staging row stride (conflict-free, even)
  float* sWT1 = smem;
  float* sB1  = smem + oB1;
  float* sWT2 = smem + oW2;
  float* sB2  = smem + oB2;
  float* sWT3 = smem + oW3;
  float* sB3  = smem + oB3;
  const int tid = threadIdx.x;

  // ---- stage transposed weights (coalesced global reads) ----
  for (int i = tid; i < C1 * 4; i += 256) {
    int n = i >> 2, r = i & 3;
    sWT1[n * ST1 + r] = (r < 3) ? W1[r * C1 + n] : 0.f;   // pad k=3 row
  }
  for (int i = tid; i < C1; i += 256) sB1[i] = B1[i];
  for (int i = tid; i < C1 * C2; i += 256) {
    int r = i / C2, n = i % C2;
    sWT2[n * ST2 + r] = W2[i];
  }
  for (int i = tid; i < C2; i += 256) sB2[i] = B2[i];
  for (int i = tid; i < C2 * C3; i += 256) {
    int r = i / C3, n = i % C3;
    sWT3[n * ST3 + r] = W3[i];
  }
  for (int i = tid; i < C3; i += 256) sB3[i] = B3[i];
  __syncthreads();

  const int lane = tid & 31;
  const int wid  = tid >> 5;
  const int l15  = lane & 15;
  const bool lo  = lane < 16;
  const int mrow = lo ? 0 : 8;
  const int cent = blockIdx.x * 8 + wid;
  const float* gp = g + (size_t)cent * K * 3;
  float* stg = smem + oST + wid * 16 * STR;

  constexpr int NT1 = C1 / 16, NT2 = C2 / 16, NT3 = C3 / 16;
  constexpr int KB2 = C1 / 4,  KB3 = C2 / 4;
  constexpr int NTILES = K / 16;

  v8f vzero = {};
  v8f mx[NT3];
#pragma unroll
  for (int nt = 0; nt < NT3; ++nt) mx[nt] = vzero;   // ReLU outputs >= 0

  // ---- hoist t-invariant fragments and biases into registers ----
  const int kr1 = lo ? 0 : 2;
  v2f bf1[NT1];
  float bv1[NT1], bv2[NT2], bv3[NT3];
#pragma unroll
  for (int nt = 0; nt < NT1; ++nt) {
    int n = nt * 16 + l15;
    bf1[nt] = *(const v2f*)(sWT1 + n * ST1 + kr1);
    bv1[nt] = sB1[n];
  }
#pragma unroll
  for (int nt = 0; nt < NT2; ++nt) bv2[nt] = sB2[nt * 16 + l15];
#pragma unroll
  for (int nt = 0; nt < NT3; ++nt) bv3[nt] = sB3[nt * 16 + l15];

#pragma clang loop unroll(disable)
  for (int t = 0; t < NTILES; ++t) {
    // ---- layer 1: (16 x 3pad4) x (4 x C1) ----
    const float* gr = gp + (t * 16 + l15) * 3;
    v2f a0;
    a0.x = lo ? gr[0] : gr[2];
    a0.y = lo ? gr[1] : 0.f;                          // padded k=3 column
#pragma unroll
    for (int nt = 0; nt < NT1; ++nt) {
      v8f d = wmma4(a0, bf1[nt], vzero);
      int n = nt * 16 + l15;
#pragma unroll
      for (int e = 0; e < 8; ++e)
        stg[(mrow + e) * STR + n] = fmaxf(d[e] + bv1[nt], 0.f);
    }

    // ---- layer 2: (16 x C1) x (C1 x C2) ----
    v8f acc[NT2];
#pragma unroll
    for (int nt = 0; nt < NT2; ++nt) acc[nt] = vzero;
#pragma unroll
    for (int kb = 0; kb < KB2; ++kb) {
      int kc = 4 * kb + (lo ? 0 : 2);
      v2f a = *(const v2f*)(stg + l15 * STR + kc);
#pragma unroll
      for (int nt = 0; nt < NT2; ++nt) {
        v2f bf = *(const v2f*)(sWT2 + (nt * 16 + l15) * ST2 + kc);
        acc[nt] = wmma4(a, bf, acc[nt]);
      }
    }
#pragma unroll
    for (int nt = 0; nt < NT2; ++nt) {
      int n = nt * 16 + l15;
#pragma unroll
      for (int e = 0; e < 8; ++e)
        stg[(mrow + e) * STR + n] = fmaxf(acc[nt][e] + bv2[nt], 0.f);
    }

    // ---- layer 3: (16 x C2) x (C2 x C3), fused ReLU + maxpool ----
    v8f acc3[NT3];
#pragma unroll
    for (int nt = 0; nt < NT3; ++nt) acc3[nt] = vzero;
#pragma unroll
    for (int kb = 0; kb < KB3; ++kb) {
      int kc = 4 * kb + (lo ? 0 : 2);
      v2f a = *(const v2f*)(stg + l15 * STR + kc);
#pragma unroll
      for (int nt = 0; nt < NT3; ++nt) {
        v2f bf = *(const v2f*)(sWT3 + (nt * 16 + l15) * ST3 + kc);
        acc3[nt] = wmma4(a, bf, acc3[nt]);
      }
    }
#pragma unroll
    for (int nt = 0; nt < NT3; ++nt) {
#pragma unroll
      for (int e = 0; e < 8; ++e) {
        float v = fmaxf(acc3[nt][e] + bv3[nt], 0.f);
        mx[nt][e] = fmaxf(mx[nt][e], v);
      }
    }
  }

  // reduce over the 8 M-rows per half, then combine lane halves (M 0-7 vs 8-15)
#pragma unroll
  for (int nt = 0; nt < NT3; ++nt) {
    float v = mx[nt][0];
#pragma unroll
    for (int e = 1; e < 8; ++e) v = fmaxf(v, mx[nt][e]);
    float o = __shfl_xor(v, 16, 32);
    v = fmaxf(v, o);
    if (lo) out[(size_t)cent * OUTC + chanoff + nt * 16 + l15] = v;
  }
}

// ---------------------------------------------------------------------------
extern "C" void kernel_launch(void* const* d_in, const int* in_sizes, int n_in,
                              void* d_out, int out_size, void* d_ws, size_t ws_size,
                              hipStream_t stream) {
  (void)in_sizes; (void)n_in; (void)out_size; (void)ws_size;
  const float* x = (const float*)d_in[0];
  const int* start_idx = (const int*)d_in[1];
  const float* W[3][3]; const float* Bb[3][3];
  for (int s = 0; s < 3; ++s)
    for (int l = 0; l < 3; ++l) {
      W[s][l]  = (const float*)d_in[2 + s * 6 + l * 2];
      Bb[s][l] = (const float*)d_in[2 + s * 6 + l * 2 + 1];
    }
  float* out = (float*)d_out;
  float* ws  = (float*)d_ws;
  float* centroids = ws;                               // 8192*3
  float* g0 = ws + (size_t)8192 * 3;                   // 8192*16*3
  float* g1 = g0 + (size_t)8192 * 16 * 3;              // 8192*32*3
  float* g2 = g1 + (size_t)8192 * 32 * 3;              // 8192*128*3

  fps_kernel<<<8, FPS_T, 0, stream>>>(x, start_idx, centroids, out);

  ballq_kernel<<<1024, 256, 0, stream>>>(x, centroids, g0, 16,  (float)(0.1 * 0.1));
  ballq_kernel<<<1024, 256, 0, stream>>>(x, centroids, g1, 32,  (float)(0.2 * 0.2));
  ballq_kernel<<<1024, 256, 0, stream>>>(x, centroids, g2, 128, (float)(0.4 * 0.4));

  auto lds_bytes = [](int C1, int C2, int C3) -> size_t {
    return (size_t)(C1 * 6 + C1 + C2 * (C1 + 4) + C2 + C3 * (C2 + 4) + C3 +
                    8 * 16 * 100) * sizeof(float);
  };
  size_t l0 = lds_bytes(32, 32, 64);
  size_t l1 = lds_bytes(64, 64, 128);
  size_t l2 = lds_bytes(64, 96, 128);
  (void)hipFuncSetAttribute(reinterpret_cast<const void*>(&mlp_kernel<16, 32, 32, 64>),
                            hipFuncAttributeMaxDynamicSharedMemorySize, (int)l0);
  (void)hipFuncSetAttribute(reinterpret_cast<const void*>(&mlp_kernel<32, 64, 64, 128>),
                            hipFuncAttributeMaxDynamicSharedMemorySize, (int)l1);
  (void)hipFuncSetAttribute(reinterpret_cast<const void*>(&mlp_kernel<128, 64, 96, 128>),
                            hipFuncAttributeMaxDynamicSharedMemorySize, (int)l2);

  mlp_kernel<16, 32, 32, 64><<<1024, 256, l0, stream>>>(
      g0, W[0][0], Bb[0][0], W[0][1], Bb[0][1], W[0][2], Bb[0][2], out, 3);
  mlp_kernel<32, 64, 64, 128><<<1024, 256, l1, stream>>>(
      g1, W[1][0], Bb[1][0], W[1][1], Bb[1][1], W[1][2], Bb[1][2], out, 67);
  mlp_kernel<128, 64, 96, 128><<<1024, 256, l2, stream>>>(
      g2, W[2][0], Bb[2][0], W[2][1], Bb[2][1], W[2][2], Bb[2][2], out, 195);
}